// FlaxLinear_10883447128700
// MI455X (gfx1250) — compile-verified
//
#include <hip/hip_runtime.h>
#include <hip/hip_bf16.h>

typedef __attribute__((ext_vector_type(2))) float v2f;
typedef __attribute__((ext_vector_type(8))) float v8f;

#define MULV 512
#define DTOT 4608                      // 512*1 + 512*3 + 512*5
#define PWEIGHT 0.044194173824159216f  // 512^-0.5

#define TM 32     // m rows per block (2 x 16 per wave)
#define TN 128    // n cols per block (4 waves x 32)
#define TK 64     // k chunk staged in LDS

#define LDA 65    // padded LDS row stride for A  (TM x TK)
#define LDB 132   // padded LDS row stride for W  (TK x TN), 132*4B = 33*16B aligned

// ---- CDNA5 async global->LDS copies (ASYNCcnt-tracked, no VGPR staging) ----
__device__ __forceinline__ void async_ld_b32(unsigned lds_off, unsigned long long gaddr) {
    asm volatile("global_load_async_to_lds_b32 %0, %1, off"
                 :: "v"(lds_off), "v"(gaddr) : "memory");
}
__device__ __forceinline__ void async_ld_b128(unsigned lds_off, unsigned long long gaddr) {
    asm volatile("global_load_async_to_lds_b128 %0, %1, off"
                 :: "v"(lds_off), "v"(gaddr) : "memory");
}
// wait until at most N async ops outstanding (loads retire in order per ISA)
__device__ __forceinline__ void wait_async32() {
    asm volatile("s_wait_asynccnt 0x20" ::: "memory");
}
__device__ __forceinline__ void wait_async0() {
    asm volatile("s_wait_asynccnt 0" ::: "memory");
}

template <int D, int OFF, bool BIAS>
__global__ __launch_bounds__(128) void e3nn_linear_wmma_f32(
    const float* __restrict__ x,
    const float* __restrict__ w,
    const float* __restrict__ b0,
    float* __restrict__ out)
{
    __shared__ float lds_a[2][TM * LDA];   // ping-pong [m][k]
    __shared__ float lds_b[2][TK * LDB];   // ping-pong [k][n]

    const int m0 = blockIdx.x * TM;
    const int n0 = blockIdx.y * TN;
    const int i  = blockIdx.z;             // 0..D-1, irrep component

    const int t      = threadIdx.x;        // 0..127
    const int lane   = t & 31;
    const int waveId = t >> 5;
    const int kHalf  = lane >> 4;          // 0: K pair {0,1}; 1: K pair {2,3}
    const int mn     = lane & 15;          // M for A-frags, N for B-frags
    const int nw     = waveId * 32;        // wave's 32-col slice

    // stage one K-chunk into buffer `buf` (32 async ops per thread)
    auto stage = [&](int kc, int buf) {
        // A chunk 32x64: per-lane strided gathers (compile-time stride D)
        #pragma unroll
        for (int j = 0; j < 16; ++j) {
            const int f = t * 16 + j;          // 2048 elems / 128 threads
            const int m = f >> 6;
            const int k = f & 63;
            async_ld_b32((unsigned)(uintptr_t)&lds_a[buf][m * LDA + k],
                         (unsigned long long)(uintptr_t)
                         (x + (size_t)(m0 + m) * DTOT + OFF + (size_t)(kc + k) * D + i));
        }
        // W chunk 64x128: 16B-vector copies (coalesced rows)
        #pragma unroll
        for (int j = 0; j < 16; ++j) {
            const int f  = t + 128 * j;        // 2048 float4 / 128 threads
            const int k  = f >> 5;             // 32 float4 per 128-col row
            const int c4 = f & 31;
            async_ld_b128((unsigned)(uintptr_t)&lds_b[buf][k * LDB + c4 * 4],
                          (unsigned long long)(uintptr_t)
                          (w + (size_t)(kc + k) * MULV + n0 + c4 * 4));
        }
    };

    v8f c00 = {}, c01 = {}, c10 = {}, c11 = {};   // 2x2 register blocking

    stage(0, 0);                     // prologue fill
    int cur = 0;
    for (int kc = 0; kc < MULV; kc += TK) {
        if (kc + TK < MULV) {
            stage(kc + TK, cur ^ 1); // prefetch next chunk into other buffer
            wait_async32();          // oldest 32 (current chunk) retired
        } else {
            wait_async0();
        }
        __syncthreads();

        // ---- 16 K-steps, 4 WMMAs each (A/B fragments reused 2x) ----
        #pragma unroll
        for (int kk = 0; kk < TK; kk += 4) {
            const int kr = kk + 2 * kHalf;
            v2f a0, a1, bA, bB;
            a0.x = lds_a[cur][mn * LDA + kr];          a0.y = lds_a[cur][mn * LDA + kr + 1];
            a1.x = lds_a[cur][(mn + 16) * LDA + kr];   a1.y = lds_a[cur][(mn + 16) * LDA + kr + 1];
            bA.x = lds_b[cur][kr * LDB + nw + mn];        bA.y = lds_b[cur][(kr + 1) * LDB + nw + mn];
            bB.x = lds_b[cur][kr * LDB + nw + 16 + mn];   bB.y = lds_b[cur][(kr + 1) * LDB + nw + 16 + mn];
            c00 = __builtin_amdgcn_wmma_f32_16x16x4_f32(false, a0, false, bA, (short)0, c00, false, false);
            c01 = __builtin_amdgcn_wmma_f32_16x16x4_f32(false, a0, false, bB, (short)0, c01, false, false);
            c10 = __builtin_amdgcn_wmma_f32_16x16x4_f32(false, a1, false, bA, (short)0, c10, false, false);
            c11 = __builtin_amdgcn_wmma_f32_16x16x4_f32(false, a1, false, bB, (short)0, c11, false, false);
        }
        __syncthreads();             // all reads done before buffer is refilled
        cur ^= 1;
    }

    // ---- epilogue: path weight, bias on scalar irrep, strided stores ----
    const int col0 = n0 + nw + mn;
    const int col1 = col0 + 16;
    const float bias0 = BIAS ? b0[col0] : 0.0f;
    const float bias1 = BIAS ? b0[col1] : 0.0f;
    #pragma unroll
    for (int r = 0; r < 8; ++r) {
        const int Mr0 = m0 + r + 8 * kHalf;   // C/D layout: VGPR r -> M=r (+8 upper half)
        const int Mr1 = Mr0 + 16;
        out[(size_t)Mr0 * DTOT + OFF + (size_t)col0 * D + i] = PWEIGHT * c00[r] + bias0;
        out[(size_t)Mr0 * DTOT + OFF + (size_t)col1 * D + i] = PWEIGHT * c01[r] + bias1;
        out[(size_t)Mr1 * DTOT + OFF + (size_t)col0 * D + i] = PWEIGHT * c10[r] + bias0;
        out[(size_t)Mr1 * DTOT + OFF + (size_t)col1 * D + i] = PWEIGHT * c11[r] + bias1;
    }
}

extern "C" void kernel_launch(void* const* d_in, const int* in_sizes, int n_in,
                              void* d_out, int out_size, void* d_ws, size_t ws_size,
                              hipStream_t stream) {
    const float* x  = (const float*)d_in[0];
    const float* w0 = (const float*)d_in[1];
    const float* w1 = (const float*)d_in[2];
    const float* w2 = (const float*)d_in[3];
    const float* b0 = (const float*)d_in[4];
    float* out = (float*)d_out;

    const dim3 blk(128);
    // irrep 0e: d=1, offset 0, bias
    e3nn_linear_wmma_f32<1, 0, true>
        <<<dim3(4096 / TM, MULV / TN, 1), blk, 0, stream>>>(x, w0, b0, out);
    // irrep 1e: d=3, offset 512
    e3nn_linear_wmma_f32<3, 512, false>
        <<<dim3(4096 / TM, MULV / TN, 3), blk, 0, stream>>>(x, w1, b0, out);
    // irrep 2e: d=5, offset 2048
    e3nn_linear_wmma_f32<5, 2048, false>
        <<<dim3(4096 / TM, MULV / TN, 5), blk, 0, stream>>>(x, w2, b0, out);
}